// SetCriterion_65120294142372
// MI455X (gfx1250) — compile-verified
//
#include <hip/hip_runtime.h>
#include <hip/hip_bf16.h>

#define B_   32
#define P_   2048
#define S_   2048
#define M_   256
#define NPTS 40000
#define C_   18
#define NEARr 0.3f
#define FARr  0.6f
#define EPSr  1e-6f

typedef float v2f __attribute__((ext_vector_type(2)));
typedef float v8f __attribute__((ext_vector_type(8)));

// ---------------------------------------------------------------- init
__global__ void init_kernel(float* acc, unsigned int* c2bits) {
    int t = blockIdx.x * blockDim.x + threadIdx.x;
    if (t < 16) acc[t] = 0.0f;
    if (t < B_ * M_) c2bits[t] = 0x7F800000u;  // +inf
}

// ------------------------------------------------- WMMA pairwise NN
// One wave (32 threads) per 16-proposal tile; loops over all 16-gt tiles.
// dist^2[m][n] = |p_m|^2 + |g_n|^2 - 2 * <p_m, g_n>, dot via V_WMMA_F32_16X16X4_F32.
__global__ void nn16_kernel(const float* __restrict__ A,     // (B,P,3)
                            const float* __restrict__ G,     // (B,M,3)
                            float* __restrict__ rowMin,      // (B,P)
                            int*   __restrict__ rowArg,      // (B,P) or null
                            unsigned int* __restrict__ colMinBits) // (B,M) or null
{
    const int b     = blockIdx.y;
    const int ptile = blockIdx.x;
    const int lane  = threadIdx.x;   // 0..31

    __shared__ float px[16], py[16], pz[16], pn2[16];
    __shared__ float gx[16], gy[16], gz[16], gn2[16];

    const float* Ab = A + ((size_t)b * P_ + (size_t)ptile * 16) * 3;
    if (lane < 16) {
        float x = Ab[lane * 3 + 0], y = Ab[lane * 3 + 1], z = Ab[lane * 3 + 2];
        px[lane] = x; py[lane] = y; pz[lane] = z;
        pn2[lane] = x * x + y * y + z * z;
    }
    __syncthreads();

    // 32-bit A-matrix 16x4 layout: lanes 0-15 -> K=0,1 ; lanes 16-31 -> K=2,3
    v2f av;
    if (lane < 16) { av.x = px[lane];      av.y = py[lane]; }
    else           { av.x = pz[lane - 16]; av.y = 0.0f;     }

    float best[8];
    int   barg[8];
#pragma unroll
    for (int r = 0; r < 8; ++r) { best[r] = 3.0e38f; barg[r] = 0; }

    const int mhi = (lane >> 4) * 8;   // row offset of this half-wave
    const int n   = lane & 15;         // column this lane owns in C/D

    for (int mt = 0; mt < M_ / 16; ++mt) {
        const float* Gb = G + ((size_t)b * M_ + (size_t)mt * 16) * 3;
        if (lane < 16) {
            float x = Gb[lane * 3 + 0], y = Gb[lane * 3 + 1], z = Gb[lane * 3 + 2];
            gx[lane] = x; gy[lane] = y; gz[lane] = z;
            gn2[lane] = x * x + y * y + z * z;
        }
        __syncthreads();

        v2f bv;
        if (lane < 16) { bv.x = gx[lane];      bv.y = gy[lane]; }
        else           { bv.x = gz[lane - 16]; bv.y = 0.0f;     }

        v8f dot = {};
        dot = __builtin_amdgcn_wmma_f32_16x16x4_f32(
            /*neg_a=*/false, av, /*neg_b=*/false, bv,
            /*c_mod=*/(short)0, dot, /*reuse_a=*/false, /*reuse_b=*/false);

        float g2 = gn2[n];
        float colmin = 3.0e38f;
#pragma unroll
        for (int r = 0; r < 8; ++r) {
            int   m = r + mhi;
            float d = fmaxf(pn2[m] + g2 - 2.0f * dot[r], 0.0f);
            int gidx = mt * 16 + n;
            if (d < best[r]) { best[r] = d; barg[r] = gidx; }
            colmin = fminf(colmin, d);
        }
        if (colMinBits) {
            float other = __shfl_xor(colmin, 16, 32);   // rows 0-7 vs 8-15, same column
            colmin = fminf(colmin, other);
            if (lane < 16)
                atomicMin(&colMinBits[(size_t)b * M_ + (size_t)mt * 16 + n],
                          __float_as_uint(colmin));
        }
        __syncthreads();
    }

    // row min+argmin across the 16 columns (lanes within each half-wave)
#pragma unroll
    for (int r = 0; r < 8; ++r) {
        float v = best[r]; int idx = barg[r];
#pragma unroll
        for (int off = 8; off >= 1; off >>= 1) {
            float v2 = __shfl_xor(v, off, 16);
            int   i2 = __shfl_xor(idx, off, 16);
            if (v2 < v || (v2 == v && i2 < idx)) { v = v2; idx = i2; }
        }
        if ((lane & 15) == 0) {
            int m = r + mhi;
            size_t o = (size_t)b * P_ + (size_t)ptile * 16 + m;
            rowMin[o] = v;
            if (rowArg) rowArg[o] = idx;
        }
    }
}

// --------------------------------------------------- block reduction
__device__ __forceinline__ float block_reduce_sum(float v) {
    __shared__ float s[32];
    int lane = threadIdx.x & 31;
    int wid  = threadIdx.x >> 5;
#pragma unroll
    for (int off = 16; off; off >>= 1) v += __shfl_down(v, off, 32);
    if (lane == 0) s[wid] = v;
    __syncthreads();
    int nw = (blockDim.x + 31) >> 5;
    v = (threadIdx.x < nw) ? s[threadIdx.x] : 0.0f;
    if (wid == 0) {
#pragma unroll
        for (int off = 16; off; off >>= 1) v += __shfl_down(v, off, 32);
    }
    __syncthreads();
    return v;  // valid in thread 0
}

// ---------------------------------------------------------- vote loss
__global__ void vote_loss_kernel(const float* __restrict__ vote_xyz,
                                 const float* __restrict__ seed_xyz,
                                 const float* __restrict__ vote_label,
                                 const int*   __restrict__ seed_inds,
                                 const int*   __restrict__ vote_label_mask,
                                 float* acc)
{
    int tid = blockIdx.x * blockDim.x + threadIdx.x;
    float num = 0.0f, den = 0.0f;
    if (tid < B_ * S_) {
        int b = tid / S_, s = tid % S_;
        int si = seed_inds[(size_t)b * S_ + s];
        float mask = (float)vote_label_mask[(size_t)b * NPTS + si];
        const float* vl = vote_label + ((size_t)b * NPTS + si) * 9;
        const float* sx = seed_xyz + ((size_t)b * S_ + s) * 3;
        const float* vx = vote_xyz + ((size_t)b * S_ + s) * 3;
        float bestd = 3.0e38f;
#pragma unroll
        for (int j = 0; j < 3; ++j) {
            float dx = fabsf(vx[0] - (vl[3 * j + 0] + sx[0]));
            float dy = fabsf(vx[1] - (vl[3 * j + 1] + sx[1]));
            float dz = fabsf(vx[2] - (vl[3 * j + 2] + sx[2]));
            bestd = fminf(bestd, dx + dy + dz);
        }
        num = bestd * mask;
        den = mask;
    }
    float rn = block_reduce_sum(num);
    float rd = block_reduce_sum(den);
    if (threadIdx.x == 0) { atomicAdd(&acc[0], rn); atomicAdd(&acc[1], rd); }
}

// ------------------------------------------------------- proposal loss
__global__ void proposal_loss_kernel(const float* __restrict__ d1,
                                     const int*   __restrict__ ind1,
                                     const float* __restrict__ c1,
                                     const float* __restrict__ obj_scores,   // (B,P,2)
                                     const float* __restrict__ sem_scores,   // (B,P,C)
                                     const float* __restrict__ size_scores,  // (B,P,C)
                                     const float* __restrict__ size_res,     // (B,P,C,3)
                                     const int*   __restrict__ gt_sem,       // (B,M)
                                     const float* __restrict__ gt_sizes,     // (B,M,3)
                                     const float* __restrict__ gt_present,   // (B,M)
                                     const float* __restrict__ mean_size,    // (C,3)
                                     float* acc)
{
    int tid = blockIdx.x * blockDim.x + threadIdx.x;
    float v_ceobj = 0, v_objmask = 0, v_cesem = 0, v_objf = 0,
          v_c1 = 0, v_cesize = 0, v_hl = 0;
    if (tid < B_ * P_) {
        size_t bp = (size_t)tid;
        int b = tid / P_;
        float eu = sqrtf(fmaxf(d1[bp], 0.0f) + EPSr);
        int   lbl  = (eu < NEARr) ? 1 : 0;
        float objf = (float)lbl;
        float omsk = ((eu < NEARr) || (eu > FARr)) ? 1.0f : 0.0f;

        // weighted 2-class CE
        {
            float l0 = obj_scores[bp * 2 + 0], l1 = obj_scores[bp * 2 + 1];
            float mx = fmaxf(l0, l1);
            float lse = mx + logf(expf(l0 - mx) + expf(l1 - mx));
            float w = lbl ? 0.8f : 0.2f;
            float sel = lbl ? l1 : l0;
            v_ceobj = (lse - sel) * w * omsk;
        }

        int ind = ind1[bp];
        int cls = gt_sem[(size_t)b * M_ + ind];

        // 18-class CE for sem and size heads
        {
            const float* sl = sem_scores + bp * C_;
            const float* zl = size_scores + bp * C_;
            float mx1 = sl[0], mx2 = zl[0];
            for (int c = 1; c < C_; ++c) { mx1 = fmaxf(mx1, sl[c]); mx2 = fmaxf(mx2, zl[c]); }
            float s1 = 0.0f, s2 = 0.0f;
            for (int c = 0; c < C_; ++c) { s1 += expf(sl[c] - mx1); s2 += expf(zl[c] - mx2); }
            v_cesem  = (mx1 + logf(s1) - sl[cls]) * objf;
            v_cesize = (mx2 + logf(s2) - zl[cls]) * objf;
        }

        v_c1 = c1[bp] * objf;

        // huber on size residuals
        {
            float pres = gt_present[(size_t)b * M_ + ind];
            const float* gs = gt_sizes + ((size_t)b * M_ + ind) * 3;
            const float* ms = mean_size + (size_t)cls * 3;
            const float* pr = size_res + (bp * C_ + cls) * 3;
            float hsum = 0.0f;
#pragma unroll
            for (int k = 0; k < 3; ++k) {
                float rl = (pres == 0.0f) ? 0.0f : (gs[k] - ms[k]);
                float x  = pr[k] - rl / ms[k];
                float a  = fabsf(x);
                float q  = fminf(a, 1.0f);
                hsum += 0.5f * q * q + (a - q);
            }
            v_hl = (hsum * (1.0f / 3.0f)) * objf;
        }
        v_objmask = omsk;
        v_objf = objf;
    }
    float r;
    r = block_reduce_sum(v_ceobj);   if (threadIdx.x == 0) atomicAdd(&acc[2], r);
    r = block_reduce_sum(v_objmask); if (threadIdx.x == 0) atomicAdd(&acc[3], r);
    r = block_reduce_sum(v_cesem);   if (threadIdx.x == 0) atomicAdd(&acc[4], r);
    r = block_reduce_sum(v_objf);    if (threadIdx.x == 0) atomicAdd(&acc[5], r);
    r = block_reduce_sum(v_c1);      if (threadIdx.x == 0) atomicAdd(&acc[6], r);
    r = block_reduce_sum(v_cesize);  if (threadIdx.x == 0) atomicAdd(&acc[7], r);
    r = block_reduce_sum(v_hl);      if (threadIdx.x == 0) atomicAdd(&acc[8], r);
}

// ------------------------------------------------------------ gt loss
__global__ void gt_loss_kernel(const unsigned int* __restrict__ c2bits,
                               const float* __restrict__ gt_present,
                               float* acc)
{
    int tid = blockIdx.x * blockDim.x + threadIdx.x;
    float num = 0.0f, den = 0.0f;
    if (tid < B_ * M_) {
        float c2 = __uint_as_float(c2bits[tid]);
        float p  = gt_present[tid];
        num = c2 * p;
        den = p;
    }
    float rn = block_reduce_sum(num);
    float rd = block_reduce_sum(den);
    if (threadIdx.x == 0) { atomicAdd(&acc[9], rn); atomicAdd(&acc[10], rd); }
}

// ------------------------------------------------------------ finalize
__global__ void finalize_kernel(const float* __restrict__ acc, float* out) {
    if (threadIdx.x == 0 && blockIdx.x == 0) {
        float loss_vote   = acc[0] / (acc[1] + EPSr);
        float loss_obj    = acc[2] / (acc[3] + EPSr);
        float loss_sem    = acc[4] / (acc[5] + EPSr);
        float loss_center = acc[6] / (acc[5] + EPSr) + acc[9] / (acc[10] + EPSr);
        float size_cls    = acc[7] / (acc[5] + EPSr);
        float loss_size   = 0.1f * size_cls + acc[8] / (acc[5] + EPSr);
        out[0] = 10.0f * (loss_vote + 0.5f * loss_obj + 0.1f * loss_sem +
                          loss_center + loss_size);
    }
}

// ---------------------------------------------------------------------
extern "C" void kernel_launch(void* const* d_in, const int* in_sizes, int n_in,
                              void* d_out, int out_size, void* d_ws, size_t ws_size,
                              hipStream_t stream) {
    const float* agg_xyz   = (const float*)d_in[0];
    const float* obj_sc    = (const float*)d_in[1];
    const float* seed_xyz  = (const float*)d_in[2];
    const float* vote_xyz  = (const float*)d_in[3];
    const float* center    = (const float*)d_in[4];
    const float* size_sc   = (const float*)d_in[5];
    const float* size_res  = (const float*)d_in[6];
    const float* sem_sc    = (const float*)d_in[7];
    const float* gt_cent   = (const float*)d_in[8];
    const float* vote_lab  = (const float*)d_in[9];
    const float* gt_pres   = (const float*)d_in[10];
    const float* gt_sizes  = (const float*)d_in[11];
    const float* mean_size = (const float*)d_in[13];
    const int*   seed_inds = (const int*)d_in[14];
    const int*   vlab_mask = (const int*)d_in[15];
    const int*   gt_sem    = (const int*)d_in[16];
    float* out = (float*)d_out;

    // workspace layout
    char* ws = (char*)d_ws;
    float*        d1     = (float*)ws;                ws += (size_t)B_ * P_ * 4;
    int*          ind1   = (int*)ws;                  ws += (size_t)B_ * P_ * 4;
    float*        c1     = (float*)ws;                ws += (size_t)B_ * P_ * 4;
    unsigned int* c2bits = (unsigned int*)ws;         ws += (size_t)B_ * M_ * 4;
    float*        acc    = (float*)ws;

    init_kernel<<<(B_ * M_ + 255) / 256, 256, 0, stream>>>(acc, c2bits);

    dim3 nngrid(P_ / 16, B_);
    nn16_kernel<<<nngrid, 32, 0, stream>>>(agg_xyz, gt_cent, d1, ind1, nullptr);
    nn16_kernel<<<nngrid, 32, 0, stream>>>(center, gt_cent, c1, nullptr, c2bits);

    vote_loss_kernel<<<(B_ * S_ + 255) / 256, 256, 0, stream>>>(
        vote_xyz, seed_xyz, vote_lab, seed_inds, vlab_mask, acc);

    proposal_loss_kernel<<<(B_ * P_ + 255) / 256, 256, 0, stream>>>(
        d1, ind1, c1, obj_sc, sem_sc, size_sc, size_res,
        gt_sem, gt_sizes, gt_pres, mean_size, acc);

    gt_loss_kernel<<<(B_ * M_ + 255) / 256, 256, 0, stream>>>(c2bits, gt_pres, acc);

    finalize_kernel<<<1, 32, 0, stream>>>(acc, out);
}